// ManualLSTM_21131239097236
// MI455X (gfx1250) — compile-verified
//
#include <hip/hip_runtime.h>
#include <math.h>

#define INPUT_SIZE 4096
#define HIDDEN_SIZE 4096
#define OUTPUT_SIZE 32000
#define IH (INPUT_SIZE + HIDDEN_SIZE)

typedef __attribute__((ext_vector_type(2))) float v2f;
typedef __attribute__((ext_vector_type(8))) float v8f;

// One wave computes 16 consecutive rows of a row-major GEMV  out = W(rows x K) * v.
// WMMA roles:  A[m][k] = v[k] (broadcast over M)   B[k][n] = W[row0+n][k]
// => D[m][n] = dot(W[row0+n], v) for every m; result sits in acc VGPR0, lane n.
// Layouts per CDNA5 ISA (32-bit A 16x4 / B 4x16): VGPR0 = K0 (lanes0-15) | K2 (lanes16-31),
// VGPR1 = K1 | K3.  So lane L streams row row0+(L&15) at column offset (L<16 ? 0 : 2)
// with aligned 8-byte loads; the vector is a wave-uniform float4 load + cndmask.
__device__ __forceinline__ float wave_gemv16(const float* __restrict__ W,
                                             const float* __restrict__ v,
                                             int row0, int K)
{
    const int  lane = threadIdx.x & 31;
    const bool lo   = lane < 16;
    const int  koff = lo ? 0 : 2;
    const float* wp = W + (size_t)(row0 + (lane & 15)) * (size_t)K + koff;

    v8f acc0 = {};
    v8f acc1 = {};   // second accumulator breaks the WMMA RAW dependency chain

#pragma unroll 4
    for (int k = 0; k < K; k += 8) {
        float4 vk0 = *(const float4*)(v + k);
        float4 vk1 = *(const float4*)(v + k + 4);
        float2 w0  = *(const float2*)(wp + k);
        float2 w1  = *(const float2*)(wp + k + 4);

        v2f a0, a1, b0, b1;
        a0.x = lo ? vk0.x : vk0.z;  a0.y = lo ? vk0.y : vk0.w;
        a1.x = lo ? vk1.x : vk1.z;  a1.y = lo ? vk1.y : vk1.w;
        b0.x = w0.x;  b0.y = w0.y;
        b1.x = w1.x;  b1.y = w1.y;

        acc0 = __builtin_amdgcn_wmma_f32_16x16x4_f32(false, a0, false, b0,
                                                     (short)0, acc0, false, false);
        acc1 = __builtin_amdgcn_wmma_f32_16x16x4_f32(false, a1, false, b1,
                                                     (short)0, acc1, false, false);
    }
    return acc0[0] + acc1[0];   // valid in lanes 0..15 (row = row0 + lane)
}

__global__ void concat_xh(const float* __restrict__ x, const float* __restrict__ h,
                          float* __restrict__ comb)
{
    int j = blockIdx.x * blockDim.x + threadIdx.x;
    if (j < INPUT_SIZE)      comb[j] = x[j];
    else if (j < IH)         comb[j] = h[j - INPUT_SIZE];
}

// 1024 waves: 4 gates x 256 tiles x 16 rows.  Pre-activations (no bias yet) -> ws.
__global__ void lstm_gates_gemv(const float* __restrict__ Wf, const float* __restrict__ Wi,
                                const float* __restrict__ Wo, const float* __restrict__ Wc,
                                const float* __restrict__ comb, float* __restrict__ preact)
{
    int wave = (blockIdx.x * blockDim.x + threadIdx.x) >> 5;
    int gate = wave >> 8;                // 256 tiles per gate
    int row0 = (wave & 255) << 4;
    const float* W = (gate == 0) ? Wf : (gate == 1) ? Wi : (gate == 2) ? Wo : Wc;

    float r = wave_gemv16(W, comb, row0, IH);

    int lane = threadIdx.x & 31;
    if (lane < 16)
        preact[gate * HIDDEN_SIZE + row0 + lane] = r;
}

__global__ void lstm_pointwise(const float* __restrict__ preact,
                               const float* __restrict__ bf, const float* __restrict__ bi,
                               const float* __restrict__ bo, const float* __restrict__ bc,
                               const float* __restrict__ c,
                               float* __restrict__ h_new, float* __restrict__ c_new)
{
    int j = blockIdx.x * blockDim.x + threadIdx.x;
    if (j >= HIDDEN_SIZE) return;
    float gf = preact[0 * HIDDEN_SIZE + j] + bf[j];
    float gi = preact[1 * HIDDEN_SIZE + j] + bi[j];
    float go = preact[2 * HIDDEN_SIZE + j] + bo[j];
    float gc = preact[3 * HIDDEN_SIZE + j] + bc[j];
    float f  = 1.f / (1.f + expf(-gf));
    float i  = 1.f / (1.f + expf(-gi));
    float o  = 1.f / (1.f + expf(-go));
    float cc = tanhf(gc);
    float cn = f * c[j] + i * cc;
    float hn = o * tanhf(cn);
    c_new[j] = cn;
    h_new[j] = hn;
}

// 2000 waves: 32000 rows / 16.  logits -> d_out[0:32000] (bias fused).
__global__ void out_gemv(const float* __restrict__ Wout, const float* __restrict__ h,
                         const float* __restrict__ bout, float* __restrict__ logits)
{
    int wave = (blockIdx.x * blockDim.x + threadIdx.x) >> 5;
    int row0 = wave << 4;

    float r = wave_gemv16(Wout, h, row0, HIDDEN_SIZE);

    int lane = threadIdx.x & 31;
    if (lane < 16)
        logits[row0 + lane] = r + bout[row0 + lane];
}

// Single block, in-place numerically stable log_softmax over 32000 logits.
__global__ void log_softmax_inplace(float* __restrict__ logits)
{
    __shared__ float red[1024];
    int t = threadIdx.x;

    float m = -INFINITY;
    for (int j = t; j < OUTPUT_SIZE; j += 1024) m = fmaxf(m, logits[j]);
    red[t] = m;
    __syncthreads();
    for (int s = 512; s > 0; s >>= 1) {
        if (t < s) red[t] = fmaxf(red[t], red[t + s]);
        __syncthreads();
    }
    float mx = red[0];
    __syncthreads();

    float sum = 0.f;
    for (int j = t; j < OUTPUT_SIZE; j += 1024) sum += expf(logits[j] - mx);
    red[t] = sum;
    __syncthreads();
    for (int s = 512; s > 0; s >>= 1) {
        if (t < s) red[t] += red[t + s];
        __syncthreads();
    }
    float lse = mx + logf(red[0]);
    __syncthreads();

    for (int j = t; j < OUTPUT_SIZE; j += 1024) logits[j] -= lse;
}

extern "C" void kernel_launch(void* const* d_in, const int* in_sizes, int n_in,
                              void* d_out, int out_size, void* d_ws, size_t ws_size,
                              hipStream_t stream)
{
    const float* x    = (const float*)d_in[0];
    const float* h    = (const float*)d_in[1];
    const float* c    = (const float*)d_in[2];
    const float* Wf   = (const float*)d_in[3];
    const float* bf   = (const float*)d_in[4];
    const float* Wi   = (const float*)d_in[5];
    const float* bi   = (const float*)d_in[6];
    const float* Wo   = (const float*)d_in[7];
    const float* bo   = (const float*)d_in[8];
    const float* Wc   = (const float*)d_in[9];
    const float* bc   = (const float*)d_in[10];
    const float* Wout = (const float*)d_in[11];
    const float* bout = (const float*)d_in[12];

    float* out        = (float*)d_out;
    float* out_logits = out;                        // [0, 32000)
    float* out_h      = out + OUTPUT_SIZE;          // [32000, 36096)
    float* out_c      = out_h + HIDDEN_SIZE;        // [36096, 40192)

    float* comb   = (float*)d_ws;                   // 8192 floats
    float* preact = comb + IH;                      // 4*4096 floats

    concat_xh      <<<IH / 256, 256, 0, stream>>>(x, h, comb);
    lstm_gates_gemv<<<(4 * HIDDEN_SIZE / 16) / 8, 256, 0, stream>>>(Wf, Wi, Wo, Wc, comb, preact);
    lstm_pointwise <<<HIDDEN_SIZE / 256, 256, 0, stream>>>(preact, bf, bi, bo, bc, c, out_h, out_c);
    out_gemv       <<<(OUTPUT_SIZE / 16) / 8, 256, 0, stream>>>(Wout, out_h, bout, out_logits);
    log_softmax_inplace<<<1, 1024, 0, stream>>>(out_logits);
}